// FrequencyChannelAttention_77781857730609
// MI455X (gfx1250) — compile-verified
//
#include <hip/hip_runtime.h>
#include <hip/hip_bf16.h>
#include <math.h>

typedef float v2f __attribute__((ext_vector_type(2)));
typedef float v8f __attribute__((ext_vector_type(8)));

#define BATCH 16
#define CHAN  256
#define CB    32        // CHAN/8
#define HH    128
#define WW    128
#define SLICE (HH*WW)   // 16384
#define NSLICE (BATCH*CHAN) // 4096

// ---------------------------------------------------------------------------
// Kernel 1: wvec[i] = sum_h 2*cos(pi*(2i+1)*h/(2*128))  (u == v since H==W)
// Integer mod-512 reduction keeps the cosf argument in [0, 2*pi).
// ---------------------------------------------------------------------------
__global__ void fca_dct_colsum(float* __restrict__ wvec) {
    int i = threadIdx.x;          // 0..127
    if (i >= HH) return;
    const float scale = 3.14159265358979323846f / 256.0f;
    float s = 0.0f;
    int step = 2 * i + 1;
    for (int h = 0; h < HH; ++h) {
        int t = (step * h) & 511;           // cos(pi*t/256) has period 512 in t
        s += 2.0f * __cosf((float)t * scale);
    }
    wvec[i] = s;
}

// ---------------------------------------------------------------------------
// Kernel 2: freq_avg[b,c] = (1/16384) * sum_ij x[b,c,i,j]*wvec[i]*wvec[j]
// One 256-thread block per (b,c) slice; float4 streaming loads.
// ---------------------------------------------------------------------------
__global__ void fca_freq_reduce(const float* __restrict__ x,
                                const float* __restrict__ wvec,
                                float* __restrict__ freq_avg) {
    __shared__ float sw[HH];
    __shared__ float red[256];
    int tid = threadIdx.x;
    if (tid < HH) sw[tid] = wvec[tid];
    __syncthreads();

    const float4* xs = (const float4*)(x + (size_t)blockIdx.x * SLICE);
    float acc = 0.0f;
    // 4096 float4 per slice, 256 threads -> 16 iterations
    #pragma unroll 4
    for (int it = 0; it < 16; ++it) {
        int p  = it * 256 + tid;       // float4 index within slice
        int i  = p >> 5;               // row (32 float4 per 128-wide row)
        int j0 = (p & 31) << 2;        // starting column
        float4 v = xs[p];
        float rowsum = v.x * sw[j0]     + v.y * sw[j0 + 1]
                     + v.z * sw[j0 + 2] + v.w * sw[j0 + 3];
        acc = fmaf(rowsum, sw[i], acc);
    }
    red[tid] = acc;
    __syncthreads();
    for (int s = 128; s > 0; s >>= 1) {
        if (tid < s) red[tid] += red[tid + s];
        __syncthreads();
    }
    if (tid == 0) freq_avg[blockIdx.x] = red[0] * (1.0f / (float)SLICE);
}

// ---------------------------------------------------------------------------
// Kernel 3: MLP on the CDNA5 matrix pipe (V_WMMA_F32_16X16X4_F32).
//   h = relu(freq_avg(16x256) @ w1(256x32) + b1)
//   weights = sigmoid(h(16x32) @ w2(32x256) + b2)
// One block of 512 threads = 16 wave32s.
// Lane mapping per ISA 16x16x4 f32:
//   A/B: vgpr r of lane L holds K = 2*(L/16)+r, row/col = L%16
//   D  : vgpr r of lane L holds (m = r + 8*(L/16), n = L%16)
// ---------------------------------------------------------------------------
__global__ void fca_mlp_wmma(const float* __restrict__ fa,   // 16 x 256
                             const float* __restrict__ w1,   // 256 x 32
                             const float* __restrict__ b1,   // 32
                             const float* __restrict__ w2,   // 32 x 256
                             const float* __restrict__ b2,   // 256
                             float* __restrict__ wts) {      // 16 x 256
    __shared__ float h_sh[BATCH * CB];   // 16 x 32
    int tid  = threadIdx.x;
    int wave = tid >> 5;
    int lane = tid & 31;
    int mn   = lane & 15;                // A row (batch) / B col within tile
    int kOff = (lane >> 4) * 2;          // K half-select

    // ---- GEMM1: 2 output tiles (N=32), K=256 -> 64 wmma steps -------------
    if (wave < 2) {
        int n0 = wave * 16;
        v8f acc = {};
        #pragma unroll 8
        for (int kb = 0; kb < 64; ++kb) {
            int k0 = kb * 4 + kOff;
            v2f a, b;
            a.x = fa[mn * CHAN + k0];
            a.y = fa[mn * CHAN + k0 + 1];
            b.x = w1[(k0)     * CB + n0 + mn];
            b.y = w1[(k0 + 1) * CB + n0 + mn];
            acc = __builtin_amdgcn_wmma_f32_16x16x4_f32(
                false, a, false, b, (short)0, acc, false, false);
        }
        #pragma unroll
        for (int r = 0; r < 8; ++r) {
            int m   = r + 8 * (lane >> 4);
            int col = n0 + mn;
            float v = acc[r] + b1[col];
            h_sh[m * CB + col] = fmaxf(v, 0.0f);
        }
    }
    __syncthreads();

    // ---- GEMM2: 16 output tiles (N=256), K=32 -> 8 wmma steps each --------
    {
        int n0 = wave * 16;
        v8f acc = {};
        #pragma unroll
        for (int kb = 0; kb < 8; ++kb) {
            int k0 = kb * 4 + kOff;
            v2f a, b;
            a.x = h_sh[mn * CB + k0];
            a.y = h_sh[mn * CB + k0 + 1];
            b.x = w2[(k0)     * CHAN + n0 + mn];
            b.y = w2[(k0 + 1) * CHAN + n0 + mn];
            acc = __builtin_amdgcn_wmma_f32_16x16x4_f32(
                false, a, false, b, (short)0, acc, false, false);
        }
        #pragma unroll
        for (int r = 0; r < 8; ++r) {
            int m   = r + 8 * (lane >> 4);   // batch index
            int col = n0 + mn;               // channel index
            float v = acc[r] + b2[col];
            wts[m * CHAN + col] = 1.0f / (1.0f + __expf(-v));
        }
    }
}

// ---------------------------------------------------------------------------
// Kernel 4: out[b,c,:,:] = x[b,c,:,:] * weights[b,c]   (pure bandwidth)
// ---------------------------------------------------------------------------
__global__ void fca_scale(const float* __restrict__ x,
                          const float* __restrict__ wts,
                          float* __restrict__ out) {
    int slice = blockIdx.x;
    float w = wts[slice];
    const float4* xi = (const float4*)(x   + (size_t)slice * SLICE);
    float4*       oo = (float4*)      (out + (size_t)slice * SLICE);
    int tid = threadIdx.x;
    #pragma unroll 4
    for (int it = 0; it < 16; ++it) {
        int p = it * 256 + tid;
        float4 v = xi[p];
        v.x *= w; v.y *= w; v.z *= w; v.w *= w;
        oo[p] = v;
    }
}

// ---------------------------------------------------------------------------
extern "C" void kernel_launch(void* const* d_in, const int* in_sizes, int n_in,
                              void* d_out, int out_size, void* d_ws, size_t ws_size,
                              hipStream_t stream) {
    const float* x  = (const float*)d_in[0];  // 16*256*128*128
    const float* w1 = (const float*)d_in[1];  // 256*32
    const float* b1 = (const float*)d_in[2];  // 32
    const float* w2 = (const float*)d_in[3];  // 32*256
    const float* b2 = (const float*)d_in[4];  // 256
    float* out = (float*)d_out;

    float* ws    = (float*)d_ws;
    float* wvec  = ws;                  // 128 floats
    float* favg  = ws + 128;            // 4096 floats
    float* wts   = ws + 128 + NSLICE;   // 4096 floats

    fca_dct_colsum<<<1, 128, 0, stream>>>(wvec);
    fca_freq_reduce<<<NSLICE, 256, 0, stream>>>(x, wvec, favg);
    fca_mlp_wmma<<<1, 512, 0, stream>>>(favg, w1, b1, w2, b2, wts);
    fca_scale<<<NSLICE, 256, 0, stream>>>(x, wts, out);
}